// Encoder_28140625723761
// MI455X (gfx1250) — compile-verified
//
#include <hip/hip_runtime.h>

#define N_NODES 50000
#define N_EDGES 400000
#define VIEWS   3
#define DIM     128
#define FEAT    384   // VIEWS * DIM
#define NTILES  (N_NODES / 16)   // 3125, exact

typedef __attribute__((ext_vector_type(2))) float v2f;
typedef __attribute__((ext_vector_type(8))) float v8f;

// One wave computes a 16x128 output strip: C[tile*16 .. +16) x [v*128 .. +128)
//   C = op(A_tile) @ B_v   with op = relu(x + bias) when FUSED_RELU.
// A is (N, 384) row-major; per-view slice starts at column v*128.
// B_v = B + v*b_view_stride, row-major (128 x 128).
// WMMA f32 16x16x4 fragment layouts (ISA 7.12.2, 32-bit operands, wave32):
//   A (16x4): lane l -> m = l&15; VGPR{0,1} = A[m][k0 + 2*(l>>4) + {0,1}]
//   B (4x16): lane l -> n = l&15; VGPR{0,1} = B[k0 + 2*(l>>4) + {0,1}][n]
//   C (16x16): lane l, VGPR r -> C[r + 8*(l>>4)][l&15]
template <bool FUSED_RELU>
__global__ __launch_bounds__(256) void gemm16_wmma(
    const float* __restrict__ A,
    const float* __restrict__ B,
    const float* __restrict__ bias,   // used only when FUSED_RELU
    float* __restrict__ C,
    int b_view_stride)
{
  int wave = blockIdx.x * (blockDim.x >> 5) + (threadIdx.x >> 5);
  int lane = threadIdx.x & 31;
  if (wave >= NTILES * VIEWS) return;   // wave-uniform exit; EXEC stays all-1s

  int v    = wave % VIEWS;
  int tile = wave / VIEWS;
  int m    = lane & 15;
  int n    = lane & 15;
  int kh   = (lane >> 4) << 1;          // 0 or 2

  const float* Arow = A + (size_t)(tile * 16 + m) * FEAT + v * DIM;
  const float* Bv   = B + (size_t)v * b_view_stride;
  const float* brow = FUSED_RELU ? (bias + v * DIM) : nullptr;

  v8f acc[8] = {};   // 8 column tiles of 16

  for (int k0 = 0; k0 < DIM; k0 += 4) {
    float ax = Arow[k0 + kh];
    float ay = Arow[k0 + kh + 1];
    if (FUSED_RELU) {   // compile-time: no exec-mask branch in the hot loop
      ax = fmaxf(ax + brow[k0 + kh],     0.0f);
      ay = fmaxf(ay + brow[k0 + kh + 1], 0.0f);
    }
    v2f a; a.x = ax; a.y = ay;

    const float* b0 = Bv + (size_t)(k0 + kh) * DIM + n;
#pragma unroll
    for (int nt = 0; nt < 8; ++nt) {
      v2f b;
      b.x = b0[nt * 16];         // row k0+kh
      b.y = b0[nt * 16 + DIM];   // row k0+kh+1
      acc[nt] = __builtin_amdgcn_wmma_f32_16x16x4_f32(
          false, a, false, b, (short)0, acc[nt], false, false);
    }
  }

  int rbase = tile * 16 + ((lane >> 4) << 3);
#pragma unroll
  for (int nt = 0; nt < 8; ++nt) {
#pragma unroll
    for (int r = 0; r < 8; ++r) {
      C[(size_t)(rbase + r) * FEAT + v * DIM + nt * 16 + n] = acc[nt][r];
    }
  }
}

// out[dst[e]][:] += X[src[e]][:]  — block per edge, 384 threads (12 waves).
// Native f32 atomics resolve in L2; the 76.8MB activation buffer fits in the
// 192MB global L2, so the random gather/scatter stays on-chip.
__global__ __launch_bounds__(FEAT) void edge_agg(
    const float* __restrict__ X,
    const int* __restrict__ src,
    const int* __restrict__ dst,
    float* __restrict__ out)
{
  int e = blockIdx.x;
  int f = threadIdx.x;
  int s = src[e];
  int d = dst[e];
  float val = X[(size_t)s * FEAT + f];
  unsafeAtomicAdd(&out[(size_t)d * FEAT + f], val);
}

// out = relu(out + b2[h]);  h = feat % 128 == i & 127 since 128 | 384.
__global__ __launch_bounds__(256) void bias_relu(
    float* __restrict__ out, const float* __restrict__ b2)
{
  size_t i = (size_t)blockIdx.x * 256 + threadIdx.x;
  if (i < (size_t)N_NODES * FEAT) {
    out[i] = fmaxf(out[i] + b2[i & (DIM - 1)], 0.0f);
  }
}

extern "C" void kernel_launch(void* const* d_in, const int* in_sizes, int n_in,
                              void* d_out, int out_size, void* d_ws, size_t ws_size,
                              hipStream_t stream) {
  (void)in_sizes; (void)n_in; (void)out_size; (void)ws_size;

  const float* h   = (const float*)d_in[0];
  const int*   src = (const int*)  d_in[1];
  const int*   dst = (const int*)  d_in[2];
  const float* W1  = (const float*)d_in[3];   // (3,128,128)
  const float* b1  = (const float*)d_in[4];   // (3,128) -> flat (384)
  const float* W2  = (const float*)d_in[5];   // (128,128)
  const float* b2  = (const float*)d_in[6];   // (128)
  float* out = (float*)d_out;

  const size_t elems = (size_t)N_NODES * FEAT;   // 19.2M floats
  float* buf0 = (float*)d_ws;          // xw / yw   (76.8 MB)
  float* buf1 = buf0 + elems;          // y accumulator (76.8 MB)

  const int waves  = NTILES * VIEWS;              // 9375
  const int blocks = (waves + 7) / 8;             // 8 waves / 256-thread block

  // layer 1: xw = X_v @ W1_v ; y_pre = scatter-add over edges
  hipMemsetAsync(buf1, 0, elems * sizeof(float), stream);
  gemm16_wmma<false><<<blocks, 256, 0, stream>>>(h, W1, nullptr, buf0, DIM * DIM);
  edge_agg<<<N_EDGES, FEAT, 0, stream>>>(buf0, src, dst, buf1);

  // layer 2: yw = relu(y_pre + b1) @ W2 (bias+relu fused into GEMM A-load)
  hipMemsetAsync(out, 0, elems * sizeof(float), stream);
  gemm16_wmma<true><<<blocks, 256, 0, stream>>>(buf1, W2, b1, buf0, 0);
  edge_agg<<<N_EDGES, FEAT, 0, stream>>>(buf0, src, dst, out);

  // z = relu(z_pre + b2)
  bias_relu<<<(int)((elems + 255) / 256), 256, 0, stream>>>(out, b2);
}